// ParametricBiquad_28810640621775
// MI455X (gfx1250) — compile-verified
//
#include <hip/hip_runtime.h>

// ---------------------------------------------------------------------------
// Parametric biquad (DF2T) on MI455X / gfx1250.
//
// Strategy: linear-recurrence block decomposition.
//   state s in R^2:  s' = A s + Bv*x,  y = s1 + b0*x
//   A = [[-a1, 1], [-a2, 0]],  Bv = [b1 - a1*b0, b2 - a2*b0]
// P0: per-channel constants (h[16], p1[16], p2[16], A^1024)
// P1: zero-init scan per 1024-sample superchunk -> end states d_sc
// P2: per-channel affine combine over superchunks -> true init states S_sc
// P3: rescan with true init, emit exact state at every 16-sample tile
// P4: WMMA: for each 16-tile group, Y(16x16) = L@X + P@S via five
//     v_wmma_f32_16x16x4_f32 accumulations (L = Toeplitz(h), per-channel).
// ---------------------------------------------------------------------------

typedef float v2f __attribute__((ext_vector_type(2)));
typedef float v8f __attribute__((ext_vector_type(8)));

#define B_CH            64
#define L_LEN           262144
#define SC_LEN          1024
#define N_SC            (L_LEN / SC_LEN)      /* 256  superchunks / channel  */
#define TILES_PER_SC    (SC_LEN / 16)         /* 64   tiles / superchunk     */
#define TILES_PER_CH    (L_LEN / 16)          /* 16384 tiles / channel       */
#define GROUPS_PER_CH   (L_LEN / 256)         /* 1024 16-tile groups / chan  */
#define GROUPS_PER_WAVE 8
#define WAVES_PER_CH    (GROUPS_PER_CH / GROUPS_PER_WAVE)  /* 128 */

// Per-channel constant block (64 floats): [0..15]=h, [16..31]=p1, [32..47]=p2,
// [48..51]=A^1024 (m00,m01,m10,m11), [52]=Bv1, [53]=Bv2, [54]=a1, [55]=a2.

// ---------------------------------------------------------------- P0 -------
__global__ void bq_consts(const float* __restrict__ coeffs, float* __restrict__ C)
{
    const int ch = threadIdx.x;
    if (ch >= B_CH) return;
    const float b0 = coeffs[ch * 5 + 0];
    const float b1 = coeffs[ch * 5 + 1];
    const float b2 = coeffs[ch * 5 + 2];
    const float a1 = coeffs[ch * 5 + 3];
    const float a2 = coeffs[ch * 5 + 4];
    const float Bv1 = b1 - a1 * b0;
    const float Bv2 = b2 - a2 * b0;
    float* c = C + ch * 64;

    // impulse response h[k] = b0 (k=0), e1^T A^{k-1} Bv (k>=1)
    c[0] = b0;
    float u0 = Bv1, u1 = Bv2;
    c[1] = u0;
    for (int k = 2; k < 16; ++k) {
        const float n0 = -a1 * u0 + u1;
        const float n1 = -a2 * u0;
        u0 = n0; u1 = n1;
        c[k] = u0;
    }
    // p1[i], p2[i] = first row of A^i ; then square up to A^1024
    float m00 = 1.f, m01 = 0.f, m10 = 0.f, m11 = 1.f;
    for (int i = 0; i < 16; ++i) {
        c[16 + i] = m00;
        c[32 + i] = m01;
        const float t00 = -a1 * m00 + m10;
        const float t01 = -a1 * m01 + m11;
        const float t10 = -a2 * m00;
        const float t11 = -a2 * m01;
        m00 = t00; m01 = t01; m10 = t10; m11 = t11;    // M = A @ M
    }
    for (int sq = 0; sq < 6; ++sq) {                    // A^16 -> A^1024
        const float t00 = m00 * m00 + m01 * m10;
        const float t01 = m00 * m01 + m01 * m11;
        const float t10 = m10 * m00 + m11 * m10;
        const float t11 = m10 * m01 + m11 * m11;
        m00 = t00; m01 = t01; m10 = t10; m11 = t11;
    }
    c[48] = m00; c[49] = m01; c[50] = m10; c[51] = m11;
    c[52] = Bv1; c[53] = Bv2; c[54] = a1; c[55] = a2;
}

#define BQ_STEP(xn)                                   \
    do {                                              \
        const float _t = fmaf(-a1, s1, s2) + Bv1 * (xn); \
        s2 = fmaf(-a2, s1, Bv2 * (xn));               \
        s1 = _t;                                      \
    } while (0)

// ---------------------------------------------------------------- P1 -------
__global__ void __launch_bounds__(256)
bq_scan_zero(const float* __restrict__ x, const float* __restrict__ C,
             float* __restrict__ d_sc)
{
    const int tid = blockIdx.x * blockDim.x + threadIdx.x;
    const int ch = tid >> 8;          // N_SC == 256
    const int sc = tid & (N_SC - 1);
    const float* c = C + ch * 64;
    const float Bv1 = c[52], Bv2 = c[53], a1 = c[54], a2 = c[55];
    const float* xp = x + (size_t)ch * L_LEN + (size_t)sc * SC_LEN;
    float s1 = 0.f, s2 = 0.f;
    for (int n = 0; n < SC_LEN; n += 4) {
        const float4 xv = *(const float4*)(xp + n);
        BQ_STEP(xv.x); BQ_STEP(xv.y); BQ_STEP(xv.z); BQ_STEP(xv.w);
    }
    float* d = d_sc + (size_t)(ch * N_SC + sc) * 2;
    d[0] = s1; d[1] = s2;
}

// ---------------------------------------------------------------- P2 -------
__global__ void bq_sc_prefix(const float* __restrict__ C,
                             const float* __restrict__ d_sc,
                             float* __restrict__ S_sc)
{
    const int ch = threadIdx.x;
    if (ch >= B_CH) return;
    const float* c = C + ch * 64;
    const float m00 = c[48], m01 = c[49], m10 = c[50], m11 = c[51];
    float s1 = 0.f, s2 = 0.f;
    for (int k = 0; k < N_SC; ++k) {
        float* o = S_sc + (size_t)(ch * N_SC + k) * 2;
        o[0] = s1; o[1] = s2;                       // exclusive prefix
        const float* d = d_sc + (size_t)(ch * N_SC + k) * 2;
        const float t1 = m00 * s1 + m01 * s2 + d[0];
        const float t2 = m10 * s1 + m11 * s2 + d[1];
        s1 = t1; s2 = t2;
    }
}

// ---------------------------------------------------------------- P3 -------
__global__ void __launch_bounds__(256)
bq_scan_states(const float* __restrict__ x, const float* __restrict__ C,
               const float* __restrict__ S_sc, float* __restrict__ S)
{
    const int tid = blockIdx.x * blockDim.x + threadIdx.x;
    const int ch = tid >> 8;
    const int sc = tid & (N_SC - 1);
    const float* c = C + ch * 64;
    const float Bv1 = c[52], Bv2 = c[53], a1 = c[54], a2 = c[55];
    const float* xp = x + (size_t)ch * L_LEN + (size_t)sc * SC_LEN;
    const float* si = S_sc + (size_t)(ch * N_SC + sc) * 2;
    float s1 = si[0], s2 = si[1];
    float* Sp = S + ((size_t)ch * TILES_PER_CH + (size_t)sc * TILES_PER_SC) * 2;
    for (int t = 0; t < TILES_PER_SC; ++t) {
        Sp[t * 2 + 0] = s1;                          // state at tile start
        Sp[t * 2 + 1] = s2;
        #pragma unroll
        for (int n = 0; n < 16; n += 4) {
            const float4 xv = *(const float4*)(xp + t * 16 + n);
            BQ_STEP(xv.x); BQ_STEP(xv.y); BQ_STEP(xv.z); BQ_STEP(xv.w);
        }
    }
}

// ---------------------------------------------------------------- P4 -------
// One wave per 8 groups of 16 tiles. Fragments of L (Toeplitz of h) and
// P ([p1 p2 0 0]) are loop-invariant per wave.
__global__ void __launch_bounds__(256)
bq_wmma_apply(const float* __restrict__ x, const float* __restrict__ C,
              const float* __restrict__ S, float* __restrict__ y)
{
    const int gtid = blockIdx.x * blockDim.x + threadIdx.x;
    const int wave = gtid >> 5;
    const int lane = threadIdx.x & 31;
    const int ch   = wave / WAVES_PER_CH;
    const int gset = wave % WAVES_PER_CH;
    const int row  = lane & 15;     // M index (time within tile) / N (tile)
    const int hi   = lane >> 4;     // lanes 16..31 carry K = 2,3
    const int koff = hi * 2;

    const float* c = C + ch * 64;

    // A fragments for L: aL[kk] covers K columns [4kk+koff, 4kk+koff+1]
    v2f aL[4];
    #pragma unroll
    for (int kk = 0; kk < 4; ++kk) {
        const int j0 = 4 * kk + koff;
        const int j1 = j0 + 1;
        aL[kk][0] = (j0 <= row) ? c[row - j0] : 0.f;   // L[row][j] = h[row-j]
        aL[kk][1] = (j1 <= row) ? c[row - j1] : 0.f;
    }
    v2f aP;                                            // [p1 p2 0 0]
    aP[0] = hi ? 0.f : c[16 + row];
    aP[1] = hi ? 0.f : c[32 + row];

    const size_t chBase = (size_t)ch * L_LEN;
    const int col = row;                               // tile within group

    for (int k = 0; k < GROUPS_PER_WAVE; ++k) {
        const int g = gset * GROUPS_PER_WAVE + k;
        const float* xg = x + chBase + (size_t)g * 256 + col * 16 + koff;
        __builtin_prefetch(xg + 256, 0, 0);            // global_prefetch_b8

        v2f bX[4];                                     // X[4kk+koff..+1][col]
        #pragma unroll
        for (int kk = 0; kk < 4; ++kk) {
            const float2 v = *(const float2*)(xg + 4 * kk);
            bX[kk][0] = v.x;
            bX[kk][1] = v.y;
        }
        const int tile = g * 16 + col;
        const float2 sv = *(const float2*)(S + ((size_t)ch * TILES_PER_CH + tile) * 2);
        v2f bS;
        bS[0] = hi ? 0.f : sv.x;                       // rows 2,3 of S_pad = 0
        bS[1] = hi ? 0.f : sv.y;

        v8f acc = {};
        acc = __builtin_amdgcn_wmma_f32_16x16x4_f32(false, aP, false, bS,
                                                    (short)0, acc, false, false);
        #pragma unroll
        for (int kk = 0; kk < 4; ++kk)
            acc = __builtin_amdgcn_wmma_f32_16x16x4_f32(false, aL[kk], false, bX[kk],
                                                        (short)0, acc, false, false);

        // D layout: lane<16 -> rows 0..7 of column `col`; lane>=16 -> rows 8..15
        float* yp = y + chBase + (size_t)g * 256 + col * 16 + hi * 8;
        float4 lo, hi4;
        lo.x = acc[0]; lo.y = acc[1]; lo.z = acc[2]; lo.w = acc[3];
        hi4.x = acc[4]; hi4.y = acc[5]; hi4.z = acc[6]; hi4.w = acc[7];
        *(float4*)(yp + 0) = lo;
        *(float4*)(yp + 4) = hi4;
    }
}

// ---------------------------------------------------------------------------
extern "C" void kernel_launch(void* const* d_in, const int* in_sizes, int n_in,
                              void* d_out, int out_size, void* d_ws, size_t ws_size,
                              hipStream_t stream)
{
    const float* x      = (const float*)d_in[0];   // (64, 262144) f32
    const float* coeffs = (const float*)d_in[1];   // (64, 5)      f32
    float* y = (float*)d_out;

    // Workspace partition (floats): ~8.7 MB total.
    float* wsf   = (float*)d_ws;
    float* C     = wsf;                                   // 64*64        = 4096
    float* S     = C + (size_t)B_CH * 64;                 // 64*16384*2   = 2097152
    float* d_sc  = S + (size_t)B_CH * TILES_PER_CH * 2;   // 64*256*2     = 32768
    float* S_sc  = d_sc + (size_t)B_CH * N_SC * 2;        // 64*256*2     = 32768

    bq_consts<<<1, 64, 0, stream>>>(coeffs, C);
    bq_scan_zero<<<(B_CH * N_SC) / 256, 256, 0, stream>>>(x, C, d_sc);
    bq_sc_prefix<<<1, 64, 0, stream>>>(C, d_sc, S_sc);
    bq_scan_states<<<(B_CH * N_SC) / 256, 256, 0, stream>>>(x, C, S_sc, S);

    const int p4_threads = B_CH * WAVES_PER_CH * 32;      // 8192 waves
    bq_wmma_apply<<<p4_threads / 256, 256, 0, stream>>>(x, C, S, y);
}